// VectorQuantization_63926293234067
// MI455X (gfx1250) — compile-verified
//
#include <hip/hip_runtime.h>

#define K_CODES 1024
#define D_DIM   256
#define N_POS   32768      // 32 * 32 * 32 positions
#define HW      1024       // 32 * 32
#define BETA    0.25f
#define ROWS_PB 32         // positions (rows) per block = 2 WMMA n-tiles

typedef float v2f __attribute__((ext_vector_type(2)));
typedef float v8f __attribute__((ext_vector_type(8)));

#if __has_builtin(__builtin_amdgcn_global_load_async_to_lds_b32)
#define HAVE_ASYNC_LDS 1
typedef __attribute__((address_space(1))) int as1_int;  // global
typedef __attribute__((address_space(3))) int as3_int;  // LDS
#endif

__device__ __forceinline__ void wait_async_zero() {
#if __has_builtin(__builtin_amdgcn_s_wait_asynccnt)
    __builtin_amdgcn_s_wait_asynccnt(0);
#else
    asm volatile("s_wait_asynccnt 0x0" ::: "memory");
#endif
}

// ---------------------------------------------------------------------------
// Kernel 1: codebook row norms ||W_k||^2, one wave32 per code.
// ---------------------------------------------------------------------------
__global__ __launch_bounds__(256)
void vq_cbnorm_kernel(const float* __restrict__ cb, float* __restrict__ cbn) {
    const int wave = (blockIdx.x * blockDim.x + threadIdx.x) >> 5;  // code id
    const int lane = threadIdx.x & 31;
    if (wave >= K_CODES) return;
    const float* row = cb + (size_t)wave * D_DIM + lane * 8;
    float s = 0.f;
#pragma unroll
    for (int i = 0; i < 8; ++i) { float v = row[i]; s += v * v; }
#pragma unroll
    for (int off = 16; off >= 1; off >>= 1)
        s += __shfl_xor(s, off, 32);
    if (lane == 0) cbn[wave] = s;
}

// ---------------------------------------------------------------------------
// Kernel 2: 32 positions (2 WMMA n-tiles) per 256-thread block; each codebook
// B fragment loaded once from L2 feeds TWO fp32 WMMAs (halves L2 traffic).
// Wave w covers codes [w*128, w*128+128) for both tiles.
// ---------------------------------------------------------------------------
__global__ __launch_bounds__(256)
void vq_main_kernel(const float* __restrict__ z, const float* __restrict__ cb,
                    const float* __restrict__ cbn, float* __restrict__ zq_out,
                    float* __restrict__ idx_out, float* __restrict__ partial) {
    // two z tiles, pair-interleaved: (tile,row,d) -> tile*4096 + (d>>1)*32 + row*2 + (d&1)
    __shared__ float zt[2 * D_DIM * 16];    // 32 KB
    __shared__ float sScore[32 * 128];      // 16 KB
    __shared__ int   sCand[32 * 128];       // 16 KB
    __shared__ int   sIdx[32];
    __shared__ float sRed[256];

    const int tid   = threadIdx.x;
    const int lane  = tid & 31;
    const int wave  = tid >> 5;
    const int nbase = blockIdx.x * ROWS_PB; // 32 | 1024 => never crosses batch b
    const int b     = nbase >> 10;
    const int hw    = nbase & 1023;
    const float* zb = z + (size_t)b * (D_DIM * HW) + hw;

    // ---- stage both z tiles into LDS (32 x 256 fp32), NCHW strided gather ----
    {
        const int row32 = tid & 31;         // hw offset within block
        const int dbase = tid >> 5;         // 0..7
        const int tile  = row32 >> 4;
        const int r     = row32 & 15;
#pragma unroll
        for (int i = 0; i < 32; ++i) {
            const int d   = dbase + i * 8;  // covers d = 0..255
            const int ldi = tile * 4096 + (d >> 1) * 32 + r * 2 + (d & 1);
#ifdef HAVE_ASYNC_LDS
            __builtin_amdgcn_global_load_async_to_lds_b32(
                (as1_int*)(zb + (size_t)d * HW + row32),
                (as3_int*)&zt[ldi], 0, 0);
#else
            zt[ldi] = zb[(size_t)d * HW + row32];
#endif
        }
    }
#ifdef HAVE_ASYNC_LDS
    wait_async_zero();
#endif
    __syncthreads();

    // ---- fp32 WMMA distance GEMM, one B fragment -> two tiles ----
    const int halfsel = lane >> 4;          // fp32 A/B layout: 0 -> K=0,1; 1 -> K=2,3
    const int l16     = lane & 15;

    float best[2][8];
    int   bidx[2][8];
#pragma unroll
    for (int tt = 0; tt < 2; ++tt)
#pragma unroll
        for (int j = 0; j < 8; ++j) { best[tt][j] = 3.4e38f; bidx[tt][j] = 0; }

    for (int t = 0; t < 8; ++t) {
        const int ct  = wave * 8 + t;       // column tile 0..63
        const int col = ct * 16 + l16;
        v8f acc0 = {0.f, 0.f, 0.f, 0.f, 0.f, 0.f, 0.f, 0.f};
        v8f acc1 = {0.f, 0.f, 0.f, 0.f, 0.f, 0.f, 0.f, 0.f};
        const float* bptr  = cb + (size_t)col * D_DIM + 2 * halfsel;
        const float* a0ptr = zt + halfsel * 32 + l16 * 2;
        const float* a1ptr = a0ptr + 4096;
#pragma unroll 4
        for (int ks = 0; ks < 64; ++ks) {   // K-dim: 256 = 64 * 4
            v2f bb = *(const v2f*)(bptr + ks * 4);     // global b64 (L2)
            v2f a0 = *(const v2f*)(a0ptr + ks * 64);   // LDS b64, tile 0
            v2f a1 = *(const v2f*)(a1ptr + ks * 64);   // LDS b64, tile 1
            acc0 = __builtin_amdgcn_wmma_f32_16x16x4_f32(
                       false, a0, false, bb, (short)0, acc0, false, false);
            acc1 = __builtin_amdgcn_wmma_f32_16x16x4_f32(
                       false, a1, false, bb, (short)0, acc1, false, false);
        }
        const float cn = cbn[col];
#pragma unroll
        for (int j = 0; j < 8; ++j) {
            const float s0 = cn - 2.f * acc0[j];
            if (s0 < best[0][j]) { best[0][j] = s0; bidx[0][j] = col; }
            const float s1 = cn - 2.f * acc1[j];
            if (s1 < best[1][j]) { best[1][j] = s1; bidx[1][j] = col; }
        }
    }

    // ---- dump candidates: row = tt*16 + j + 8*halfsel, 128 cands/row ----
    {
        const int cand = wave * 16 + l16;
#pragma unroll
        for (int tt = 0; tt < 2; ++tt)
#pragma unroll
            for (int j = 0; j < 8; ++j) {
                const int row = tt * 16 + j + 8 * halfsel;
                sScore[row * 128 + cand] = best[tt][j];
                sCand[row * 128 + cand]  = bidx[tt][j];
            }
    }
    __syncthreads();

    if (tid < 32) {                         // final argmin per row (tie -> low idx)
        float bs = 3.4e38f; int bi = 0x7FFFFFFF;
        for (int c = 0; c < 128; ++c) {
            const float s = sScore[tid * 128 + c];
            const int   i = sCand[tid * 128 + c];
            if (s < bs || (s == bs && i < bi)) { bs = s; bi = i; }
        }
        sIdx[tid] = bi;
        idx_out[nbase + tid] = (float)bi;
    }
    __syncthreads();

    // ---- z_q writeback (NCHW) fused with squared-error partial ----
    float lsum = 0.f;
    {
        const int row32 = tid & 31;
        const int dbase = tid >> 5;
        const int tile  = row32 >> 4;
        const int r     = row32 & 15;
        const float* crow = cb + (size_t)sIdx[row32] * D_DIM;
        float* ob = zq_out + (size_t)b * (D_DIM * HW) + hw;
#pragma unroll
        for (int i = 0; i < 32; ++i) {
            const int d   = dbase + i * 8;
            const float q = crow[d];
            ob[(size_t)d * HW + row32] = q;
            const float zv = zt[tile * 4096 + (d >> 1) * 32 + r * 2 + (d & 1)];
            const float df = zv - q;
            lsum += df * df;
        }
    }
    sRed[tid] = lsum;
    __syncthreads();
    for (int off = 128; off >= 1; off >>= 1) {
        if (tid < off) sRed[tid] += sRed[tid + off];
        __syncthreads();
    }
    if (tid == 0) partial[blockIdx.x] = sRed[0];
}

// ---------------------------------------------------------------------------
// Kernel 3: deterministic final loss reduction over 1024 block partials.
// loss = (1 + beta) * mean((z - z_q)^2)
// ---------------------------------------------------------------------------
__global__ __launch_bounds__(256)
void vq_loss_kernel(const float* __restrict__ partial, float* __restrict__ loss) {
    __shared__ float s[256];
    float acc = 0.f;
#pragma unroll
    for (int i = 0; i < 4; ++i) acc += partial[threadIdx.x * 4 + i];
    s[threadIdx.x] = acc;
    __syncthreads();
    for (int off = 128; off >= 1; off >>= 1) {
        if (threadIdx.x < off) s[threadIdx.x] += s[threadIdx.x + off];
        __syncthreads();
    }
    if (threadIdx.x == 0)
        loss[0] = s[0] * (1.0f + BETA) / 8388608.0f;   // N_POS * D_DIM
}

// ---------------------------------------------------------------------------
extern "C" void kernel_launch(void* const* d_in, const int* in_sizes, int n_in,
                              void* d_out, int out_size, void* d_ws, size_t ws_size,
                              hipStream_t stream) {
    const float* z  = (const float*)d_in[0];   // [32, 256, 32, 32] fp32
    const float* cb = (const float*)d_in[1];   // [1024, 256] fp32

    float* out  = (float*)d_out;
    float* zq   = out;                                   // 8388608 floats (NCHW)
    float* idxf = out + (size_t)N_POS * D_DIM;           // 32768 indices (as float)
    float* loss = idxf + N_POS;                          // 1 float

    float* cbn     = (float*)d_ws;                       // 1024 floats
    float* partial = cbn + K_CODES;                      // 1024 floats

    vq_cbnorm_kernel<<<K_CODES / 8, 256, 0, stream>>>(cb, cbn);
    vq_main_kernel<<<N_POS / ROWS_PB, 256, 0, stream>>>(z, cb, cbn, zq, idxf, partial);
    vq_loss_kernel<<<1, 256, 0, stream>>>(partial, loss);
}